// GraphSAGE_1133871366801
// MI455X (gfx1250) — compile-verified
//
#include <hip/hip_runtime.h>
#include <math.h>

typedef __attribute__((ext_vector_type(2))) float v2f;
typedef __attribute__((ext_vector_type(8))) float v8f;

#define DIN  128
#define DHID 256
#define DOUT 128

// ---------------------------------------------------------------------------
// 1) zero a contiguous float region of workspace (deg | agg1 | agg2)
// ---------------------------------------------------------------------------
__global__ void zero_f32(float* __restrict__ p, long n) {
  long i = (long)blockIdx.x * blockDim.x + threadIdx.x;
  long stride = (long)gridDim.x * blockDim.x;
  for (; i < n; i += stride) p[i] = 0.0f;
}

// ---------------------------------------------------------------------------
// 2) layer-1 scatter: one wave per edge. lanes gather x[src] as float4 and
//    atomically add into agg1[dst]; lane0 accumulates degree. Atomics land in
//    L2 (agg1 + x fit in the 192MB L2), HBM only streams the edge list.
// ---------------------------------------------------------------------------
__global__ void scatter_l1(const float* __restrict__ x,
                           const int* __restrict__ ei,
                           float* __restrict__ agg,
                           float* __restrict__ deg,
                           int E) {
  int wave = blockIdx.x * (blockDim.x >> 5) + (threadIdx.x >> 5);
  int lane = threadIdx.x & 31;
  if (wave >= E) return;
  int src = ei[wave];
  int dst = ei[(long)E + wave];
  const float4* xr = (const float4*)(x + (long)src * DIN);
  float4 v = xr[lane];                       // 128 cols -> 4 per lane
  float* a = agg + (long)dst * DIN + lane * 4;
  atomicAdd(a + 0, v.x);
  atomicAdd(a + 1, v.y);
  atomicAdd(a + 2, v.z);
  atomicAdd(a + 3, v.w);
  if (lane == 0) atomicAdd(deg + dst, 1.0f);
}

// ---------------------------------------------------------------------------
// 4) layer-2 scatter: same, 256 cols -> 8 per lane (2x float4)
// ---------------------------------------------------------------------------
__global__ void scatter_l2(const float* __restrict__ h,
                           const int* __restrict__ ei,
                           float* __restrict__ agg,
                           int E) {
  int wave = blockIdx.x * (blockDim.x >> 5) + (threadIdx.x >> 5);
  int lane = threadIdx.x & 31;
  if (wave >= E) return;
  int src = ei[wave];
  int dst = ei[(long)E + wave];
  const float4* hr = (const float4*)(h + (long)src * DHID);
  float* a = agg + (long)dst * DHID;
#pragma unroll
  for (int p = 0; p < 2; ++p) {
    float4 v = hr[lane + p * 32];
    float* ap = a + (lane + p * 32) * 4;
    atomicAdd(ap + 0, v.x);
    atomicAdd(ap + 1, v.y);
    atomicAdd(ap + 2, v.z);
    atomicAdd(ap + 3, v.w);
  }
}

// ---------------------------------------------------------------------------
// 3/5) SAGE GEMM with WMMA f32 16x16x4:
//   out = act( (Aagg/deg) @ Wl  +  Aroot @ Wr  +  bias )
// One wave per 16x16 output tile; two independent WMMA accumulator chains
// so back-to-back v_wmma RAWs on C can pipeline.
// Fragment layouts per CDNA5 ISA 7.12.2:
//   A (16x4 f32):  lane M = lane%16 ; VGPR{0,1} hold K = 2*(lane/16)+{0,1}
//   B (4x16 f32):  lane N = lane%16 ; VGPR{0,1} hold K = 2*(lane/16)+{0,1}
//   C/D (16x16):   VGPR j -> M = j + 8*(lane/16), N = lane%16
// ---------------------------------------------------------------------------
template <int K, int NC, bool RELU>
__global__ void sage_gemm(const float* __restrict__ Aagg,   // [N,K] summed msgs
                          const float* __restrict__ Aroot,  // [N,K] root feats
                          const float* __restrict__ Wl,     // [K,NC]
                          const float* __restrict__ Wr,     // [K,NC]
                          const float* __restrict__ bias,   // [NC]
                          const float* __restrict__ deg,    // [N]
                          float* __restrict__ out,          // [N,NC]
                          int nTiles) {
  int wave = blockIdx.x * (blockDim.x >> 5) + (threadIdx.x >> 5);
  if (wave >= nTiles) return;                // wave-uniform: EXEC stays all-1
  const int lane = threadIdx.x & 31;
  const int nTilesN = NC / 16;
  const int m0 = (wave / nTilesN) * 16;
  const int n0 = (wave % nTilesN) * 16;
  const int half = lane >> 4;                // 0: lanes 0-15, 1: lanes 16-31
  const int row = m0 + (lane & 15);
  const int col = n0 + (lane & 15);

  const float inv = 1.0f / fmaxf(deg[row], 1.0f);   // mean = sum / max(cnt,1)
  const float* aggRow  = Aagg  + (long)row * K;
  const float* rootRow = Aroot + (long)row * K;

  v8f c0 = {};   // mean @ Wl chain
  v8f c1 = {};   // root @ Wr chain
#pragma unroll 4
  for (int k = 0; k < K; k += 4) {
    const int ka = k + half * 2;
    v2f a0, b0, a1, b1;
    a0.x = aggRow[ka] * inv;
    a0.y = aggRow[ka + 1] * inv;
    b0.x = Wl[(long)ka * NC + col];
    b0.y = Wl[(long)(ka + 1) * NC + col];
    c0 = __builtin_amdgcn_wmma_f32_16x16x4_f32(false, a0, false, b0,
                                               (short)0, c0, false, false);
    a1.x = rootRow[ka];
    a1.y = rootRow[ka + 1];
    b1.x = Wr[(long)ka * NC + col];
    b1.y = Wr[(long)(ka + 1) * NC + col];
    c1 = __builtin_amdgcn_wmma_f32_16x16x4_f32(false, a1, false, b1,
                                               (short)0, c1, false, false);
  }

  const float bv = bias[col];
#pragma unroll
  for (int j = 0; j < 8; ++j) {
    int r = m0 + j + half * 8;
    float v = c0[j] + c1[j] + bv;
    if (RELU) v = fmaxf(v, 0.0f);
    out[(long)r * NC + col] = v;
  }
}

// ---------------------------------------------------------------------------
// 6) row-wise L2 normalize in place: one wave per row of d_out (128 cols).
// ---------------------------------------------------------------------------
__global__ void l2_normalize(float* __restrict__ out, int nRows) {
  int wave = blockIdx.x * (blockDim.x >> 5) + (threadIdx.x >> 5);
  int lane = threadIdx.x & 31;
  if (wave >= nRows) return;
  float4* row = (float4*)(out + (long)wave * DOUT);
  float4 v = row[lane];                       // 128 cols -> 4 per lane
  float ss = v.x * v.x + v.y * v.y + v.z * v.z + v.w * v.w;
#pragma unroll
  for (int m = 16; m >= 1; m >>= 1) ss += __shfl_xor(ss, m, 32);
  float r = 1.0f / fmaxf(sqrtf(ss), 1e-12f);
  v.x *= r; v.y *= r; v.z *= r; v.w *= r;
  row[lane] = v;
}

// ---------------------------------------------------------------------------
// launch
// ---------------------------------------------------------------------------
extern "C" void kernel_launch(void* const* d_in, const int* in_sizes, int n_in,
                              void* d_out, int out_size, void* d_ws, size_t ws_size,
                              hipStream_t stream) {
  const float* x   = (const float*)d_in[0];
  const int*   ei  = (const int*)d_in[1];      // edge_index [2,E]
  const float* W1l = (const float*)d_in[2];
  const float* b1  = (const float*)d_in[3];
  const float* W1r = (const float*)d_in[4];
  const float* W2l = (const float*)d_in[5];
  const float* b2  = (const float*)d_in[6];
  const float* W2r = (const float*)d_in[7];
  float* out = (float*)d_out;

  const int N = in_sizes[0] / DIN;             // 50000 (multiple of 16)
  const int E = in_sizes[1] / 2;               // 1.6M

  // workspace layout: deg[N] | agg1[N*DIN] | agg2[N*DHID] | h[N*DHID]
  float* deg  = (float*)d_ws;
  float* agg1 = deg + N;
  float* agg2 = agg1 + (size_t)N * DIN;
  float* h    = agg2 + (size_t)N * DHID;

  const int BLK = 256;                         // 8 waves / block
  const int WPB = BLK / 32;

  // zero deg+agg1+agg2 (contiguous)
  long nz = (long)N * (1 + DIN + DHID);
  zero_f32<<<4096, BLK, 0, stream>>>(deg, nz);

  // layer 1: scatter -> GEMM(+bias,ReLU)
  int eBlocks = (E + WPB - 1) / WPB;
  scatter_l1<<<eBlocks, BLK, 0, stream>>>(x, ei, agg1, deg, E);

  int tiles1 = (N / 16) * (DHID / 16);
  sage_gemm<DIN, DHID, true>
      <<<(tiles1 + WPB - 1) / WPB, BLK, 0, stream>>>(agg1, x, W1l, W1r, b1,
                                                     deg, h, tiles1);

  // layer 2: scatter -> GEMM(+bias)
  scatter_l2<<<eBlocks, BLK, 0, stream>>>(h, ei, agg2, E);

  int tiles2 = (N / 16) * (DOUT / 16);
  sage_gemm<DHID, DOUT, false>
      <<<(tiles2 + WPB - 1) / WPB, BLK, 0, stream>>>(agg2, h, W2l, W2r, b2,
                                                     deg, out, tiles2);

  // row-wise L2 normalize in place
  l2_normalize<<<(N + WPB - 1) / WPB, BLK, 0, stream>>>(out, N);
}